// ChebConvMultiGraph_4054449127569
// MI455X (gfx1250) — compile-verified
//
#include <hip/hip_runtime.h>
#include <hip/hip_bf16.h>

// ChebConvMultiGraph for MI455X (gfx1250, wave32).
// Memory-bound op -> stay in fp32 and use V_WMMA_F32_16X16X4_F32 (exact fp32)
// for both GEMM stages. Intermediates live in d_ws and are L2-resident (192MB L2).
// B matrices are staged in LDS pre-packed in WMMA fragment order (one ds_load_b64
// per fragment, bank-conflict-free via stride-48 float2 padding).

#define NN   50000
#define EE   1600000
#define WW   12
#define CC   32
#define NT   3125            // NN / 16 node tiles (exact)
#define WAVES_PER_BLK 4      // 128-thread blocks = 4 wave32
#define BSTRIDE 48           // float2 stride per fragment-row: dword stride 96 == 32 (mod 64)

typedef __attribute__((ext_vector_type(2))) float v2f;
typedef __attribute__((ext_vector_type(8))) float v8f;

__device__ __forceinline__ v8f wmma_f32(v2f a, v2f b, v8f c) {
  // D = A(16x4 f32) * B(4x16 f32) + C(16x16 f32), wave32, EXEC must be all ones.
  return __builtin_amdgcn_wmma_f32_16x16x4_f32(
      /*neg_a=*/false, a, /*neg_b=*/false, b,
      /*c_mod=*/(short)0, c, /*reuse_a=*/false, /*reuse_b=*/false);
}

// ---------------------------------------------------------------- utilities
__global__ void fill_zero_kernel(float* __restrict__ p, int n) {
  int i = blockIdx.x * blockDim.x + threadIdx.x;
  int stride = gridDim.x * blockDim.x;
  for (; i < n; i += stride) p[i] = 0.0f;
}

// deg[dst] += Ew
__global__ void deg_kernel(const int* __restrict__ dst, const float* __restrict__ Ew,
                           float* __restrict__ deg) {
  int e = blockIdx.x * blockDim.x + threadIdx.x;
  if (e < EE) atomicAdd(&deg[dst[e]], Ew[e]);
}

// dinv = deg > 0 ? rsqrt(max(deg,1e-12)) : 0
__global__ void dinv_kernel(const float* __restrict__ deg, float* __restrict__ dinv) {
  int n = blockIdx.x * blockDim.x + threadIdx.x;
  if (n < NN) {
    float d = deg[n];
    dinv[n] = (d > 0.0f) ? rsqrtf(fmaxf(d, 1e-12f)) : 0.0f;
  }
}

// w_hat[e] = -dinv[src] * Ew * dinv[dst]
__global__ void what_kernel(const int* __restrict__ src, const int* __restrict__ dst,
                            const float* __restrict__ Ew, const float* __restrict__ dinv,
                            float* __restrict__ what) {
  int e = blockIdx.x * blockDim.x + threadIdx.x;
  if (e < EE) what[e] = -dinv[src[e]] * Ew[e] * dinv[dst[e]];
}

// Tx1[w,dst,c] += w_hat[e] * x[w,src,c]  : one wave per edge, lane = channel.
// 128B coalesced gather + 128B atomic scatter per (edge, w); L2-resident.
__global__ void scatter_kernel(const float* __restrict__ x,
                               const int* __restrict__ src, const int* __restrict__ dst,
                               const float* __restrict__ what,
                               float* __restrict__ tx1) {
  int wave = (blockIdx.x * blockDim.x + threadIdx.x) >> 5;
  int lane = threadIdx.x & 31;
  if (wave >= EE) return;
  int s = src[wave];
  int d = dst[wave];
  float wh = what[wave];
  #pragma unroll
  for (int w = 0; w < WW; ++w) {
    float v = x[((size_t)w * NN + s) * CC + lane] * wh;
    atomicAdd(&tx1[((size_t)w * NN + d) * CC + lane], v);
  }
}

// ------------------------------------------------- Chebyshev GEMM (WMMA f32)
// mid[w, n0:n0+16, 0:32] = [x | Tx1](16x64) @ Wcheb[w](64x32) + bcheb[w]
// Bs is fragment-packed: Bs[p*BSTRIDE + n] = {B[2p][n], B[2p+1][n]}, p = kk*2+g.
__global__ void cheb_gemm_kernel(const float* __restrict__ x,
                                 const float* __restrict__ tx1,
                                 const float* __restrict__ Wcheb,
                                 const float* __restrict__ bcheb,
                                 float* __restrict__ mid) {
  __shared__ v2f Bs[32 * BSTRIDE];              // 32 fragment-rows (K=64 / 2)
  const int w = blockIdx.y;
  const float* Bsrc = Wcheb + (size_t)w * 2048; // (kc=64) x (d=32) row-major
  for (int i = threadIdx.x; i < 1024; i += blockDim.x) {
    int p = i >> 5, n = i & 31;
    v2f t;
    t.x = Bsrc[(2 * p) * 32 + n];
    t.y = Bsrc[(2 * p + 1) * 32 + n];
    Bs[p * BSTRIDE + n] = t;
  }
  __syncthreads();

  const int tile = blockIdx.x * WAVES_PER_BLK + (threadIdx.x >> 5);
  if (tile >= NT) return;                        // wave-uniform exit
  const int lane = threadIdx.x & 31;
  const int lm = lane & 15;                      // M row / N col within tile
  const int g  = lane >> 4;                      // K-pair group
  const int n0 = tile * 16;

  const float* xrow = x   + ((size_t)w * NN + n0 + lm) * CC;
  const float* trow = tx1 + ((size_t)w * NN + n0 + lm) * CC;

  v8f acc0 = {};   // d = 0..15
  v8f acc1 = {};   // d = 16..31
  #pragma unroll
  for (int kk = 0; kk < 16; ++kk) {              // K = 64, 4 per WMMA
    const int col = kk * 4 + g * 2;
    v2f a = (col < 32) ? *(const v2f*)(xrow + col)
                       : *(const v2f*)(trow + (col - 32));
    const int p = kk * 2 + g;
    v2f b0 = Bs[p * BSTRIDE + lm];
    v2f b1 = Bs[p * BSTRIDE + 16 + lm];
    acc0 = wmma_f32(a, b0, acc0);
    acc1 = wmma_f32(a, b1, acc1);
  }

  const float bias0 = bcheb[w * CC + lm];
  const float bias1 = bcheb[w * CC + 16 + lm];
  #pragma unroll
  for (int r = 0; r < 8; ++r) {                  // C/D layout: VGPR r -> M = r + 8*g
    const int row = n0 + r + g * 8;
    float* orow = mid + ((size_t)w * NN + row) * CC;
    orow[lm]      = acc0[r] + bias0;
    orow[16 + lm] = acc1[r] + bias1;
  }
}

// ---------------------------------------- temporal conv + LeakyReLU (WMMA f32)
// out[n, w, o] = leaky( bconv[o] + sum_t mid[w+t-1](16x32) @ Wconv[:,:,t]^T(32x32) )
// Bs fragment-packed per tap: Bs[(t*16+p)*BSTRIDE + n] = {Bt[2p][n], Bt[2p+1][n]},
// where Bt[m][o] = Wconv[o][m][t].
__global__ void conv_kernel(const float* __restrict__ mid,
                            const float* __restrict__ Wconv,
                            const float* __restrict__ bconv,
                            float* __restrict__ out) {
  __shared__ v2f Bs[3 * 16 * BSTRIDE];
  for (int i = threadIdx.x; i < 3 * 512; i += blockDim.x) {
    int t = i >> 9;                 // tap 0..2
    int r = i & 511;
    int p = r >> 5, n = r & 31;
    v2f v;
    v.x = Wconv[n * 96 + (2 * p) * 3 + t];
    v.y = Wconv[n * 96 + (2 * p + 1) * 3 + t];
    Bs[(t * 16 + p) * BSTRIDE + n] = v;
  }
  __syncthreads();

  const int w = blockIdx.y;
  const int tile = blockIdx.x * WAVES_PER_BLK + (threadIdx.x >> 5);
  if (tile >= NT) return;                        // wave-uniform exit
  const int lane = threadIdx.x & 31;
  const int lm = lane & 15;
  const int g  = lane >> 4;
  const int n0 = tile * 16;

  v8f acc0 = {};
  v8f acc1 = {};
  for (int t = 0; t < 3; ++t) {                  // taps; skips are wave-uniform
    const int wt = w + t - 1;
    if (wt < 0 || wt >= WW) continue;            // zero padding
    const float* arow = mid + ((size_t)wt * NN + n0 + lm) * CC;
    const v2f* bs = Bs + t * 16 * BSTRIDE;
    #pragma unroll
    for (int kk = 0; kk < 8; ++kk) {             // K = 32
      const int col = kk * 4 + g * 2;
      v2f a = *(const v2f*)(arow + col);
      const int p = kk * 2 + g;
      v2f b0 = bs[p * BSTRIDE + lm];
      v2f b1 = bs[p * BSTRIDE + 16 + lm];
      acc0 = wmma_f32(a, b0, acc0);
      acc1 = wmma_f32(a, b1, acc1);
    }
  }

  const float bias0 = bconv[lm];
  const float bias1 = bconv[16 + lm];
  #pragma unroll
  for (int r = 0; r < 8; ++r) {
    const int row = n0 + r + g * 8;
    float v0 = acc0[r] + bias0;
    float v1 = acc1[r] + bias1;
    v0 = (v0 >= 0.0f) ? v0 : 0.01f * v0;
    v1 = (v1 >= 0.0f) ? v1 : 0.01f * v1;
    float* orow = out + ((size_t)row * WW + w) * CC;
    orow[lm]      = v0;
    orow[16 + lm] = v1;
  }
}

// -------------------------------------------------------------------- launch
extern "C" void kernel_launch(void* const* d_in, const int* in_sizes, int n_in,
                              void* d_out, int out_size, void* d_ws, size_t ws_size,
                              hipStream_t stream) {
  (void)in_sizes; (void)n_in; (void)out_size; (void)ws_size;
  const float* x     = (const float*)d_in[0];   // (W,N,32)
  const int*   A     = (const int*)  d_in[1];   // (2,E)
  const float* Ew    = (const float*)d_in[2];   // (E)
  const float* Wcheb = (const float*)d_in[3];   // (W,2,32,32)
  const float* bcheb = (const float*)d_in[4];   // (W,32)
  const float* Wconv = (const float*)d_in[5];   // (32,32,3)
  const float* bconv = (const float*)d_in[6];   // (32)
  float* out = (float*)d_out;                   // (N,W,32)

  // workspace layout (floats): deg | dinv | w_hat | Tx1 | mid   (~160 MB)
  float* ws   = (float*)d_ws;
  float* deg  = ws;
  float* dinv = ws + 50048;
  float* what = ws + 100096;
  float* tx1  = ws + 100096 + EE;               // W*N*32 floats
  float* midb = tx1 + (size_t)WW * NN * CC;

  const int* srcp = A;
  const int* dstp = A + EE;

  fill_zero_kernel<<<256, 256, 0, stream>>>(deg, NN);
  fill_zero_kernel<<<2048, 256, 0, stream>>>(tx1, WW * NN * CC);

  deg_kernel<<<(EE + 255) / 256, 256, 0, stream>>>(dstp, Ew, deg);
  dinv_kernel<<<(NN + 255) / 256, 256, 0, stream>>>(deg, dinv);
  what_kernel<<<(EE + 255) / 256, 256, 0, stream>>>(srcp, dstp, Ew, dinv, what);

  // one wave32 per edge, 8 waves per 256-thread block
  scatter_kernel<<<(EE + 7) / 8, 256, 0, stream>>>(x, srcp, dstp, what, tx1);

  dim3 gg((NT + WAVES_PER_BLK - 1) / WAVES_PER_BLK, WW);
  cheb_gemm_kernel<<<gg, WAVES_PER_BLK * 32, 0, stream>>>(x, tx1, Wcheb, bcheb, midb);
  conv_kernel<<<gg, WAVES_PER_BLK * 32, 0, stream>>>(midb, Wconv, bconv, out);
}